// TokenMixWeightGenerator_16595753632147
// MI455X (gfx1250) — compile-verified
//
#include <hip/hip_runtime.h>
#include <hip/hip_bf16.h>

#define LSEQ 2048
#define DH   64
#define NBH  16           // batch(2) * heads(8)
#define DIM  512
#define TILE 128
#define LDSS 68           // padded LDS row stride (floats): bank-conflict free
#define EPS  1e-8f

typedef __attribute__((ext_vector_type(2))) float v2f;
typedef __attribute__((ext_vector_type(8))) float v8f;

// ---------------------------------------------------------------------------
// Stage 1: per-(b,h,channel) serial scan producing
//   Cmod[bh][l][c] = C * P      (P_l = cum_A[l-1], P_0 = 1)
//   Binv[bh][l][c] = B / D      (D_l = cum_A[l-1]+eps, D_0 = 1)
// ---------------------------------------------------------------------------
__global__ void tmwg_scan_kernel(const float* __restrict__ B,
                                 const float* __restrict__ C,
                                 const float* __restrict__ S,
                                 float* __restrict__ Cmod,
                                 float* __restrict__ Binv) {
    int t = blockIdx.x * blockDim.x + threadIdx.x;
    if (t >= NBH * DH) return;
    int bh = t / DH;
    int ch = t % DH;
    int b  = bh / 8;
    int h  = bh % 8;
    const size_t inBase  = (size_t)b * LSEQ * DIM + (size_t)h * DH + ch;
    const size_t outBase = (size_t)bh * LSEQ * DH + ch;

    float run = 1.0f;                       // holds cum_A[i-1]
    for (int i = 0; i < LSEQ; ++i) {
        size_t g = inBase + (size_t)i * DIM;
        float cv = C[g];
        float bv = B[g];
        float sv = S[g];
        float P = run;
        float D = (i == 0) ? 1.0f : (run + EPS);
        size_t o = outBase + (size_t)i * DH;
        Cmod[o] = cv * P;
        Binv[o] = bv / D;
        if (i > 0) run *= sv;               // A[0] forced to 1
    }
}

// ---------------------------------------------------------------------------
// Stage 1b: diag[bh][i] = C_i . B_i  (raw, unscaled)
// ---------------------------------------------------------------------------
__global__ void tmwg_diag_kernel(const float* __restrict__ B,
                                 const float* __restrict__ C,
                                 float* __restrict__ diag) {
    int t = blockIdx.x * blockDim.x + threadIdx.x;
    if (t >= NBH * LSEQ) return;
    int bh = t / LSEQ;
    int i  = t % LSEQ;
    int b  = bh / 8;
    int h  = bh % 8;
    const float* cp = C + (size_t)b * LSEQ * DIM + (size_t)i * DIM + (size_t)h * DH;
    const float* bp = B + (size_t)b * LSEQ * DIM + (size_t)i * DIM + (size_t)h * DH;
    const float4* c4 = (const float4*)cp;
    const float4* b4 = (const float4*)bp;
    float s = 0.0f;
#pragma unroll
    for (int k = 0; k < DH / 4; ++k) {
        float4 cv = c4[k];
        float4 bv = b4[k];
        s += cv.x * bv.x + cv.y * bv.y + cv.z * bv.z + cv.w * bv.w;
    }
    diag[t] = s;
}

// ---------------------------------------------------------------------------
// Stage 2: T[bh] = Cmod @ Binv^T, strict-lower masked, diag override.
// Grid: (tj=16, ti=16, bh=16). Block: 256 threads = 8 waves.
// Each block: 128x128 tile; wave w owns rows [16w,16w+16) x all 128 cols.
// K = 64 -> 16 x V_WMMA_F32_16X16X4_F32 per 16x16 subtile.
// ---------------------------------------------------------------------------
__global__ __launch_bounds__(256)
void tmwg_gemm_kernel(const float* __restrict__ Cmod,
                      const float* __restrict__ Binv,
                      const float* __restrict__ diag,
                      float* __restrict__ out) {
    const int tj = blockIdx.x;
    const int ti = blockIdx.y;
    const int bh = blockIdx.z;
    const int tid = threadIdx.x;
    float* __restrict__ outT = out + (size_t)bh * LSEQ * LSEQ;

    // ---- strict upper tiles: zero fill (output buffer is poisoned) ----
    if (tj > ti) {
        float4 z = make_float4(0.f, 0.f, 0.f, 0.f);
#pragma unroll
        for (int it = 0; it < 16; ++it) {
            int idx = it * 256 + tid;        // 0..4095 float4 slots
            int r   = idx >> 5;              // /32 -> row 0..127
            int c   = (idx & 31) << 2;       // col (float4 granularity)
            *(float4*)(outT + (size_t)(ti * TILE + r) * LSEQ + tj * TILE + c) = z;
        }
        return;
    }

    __shared__ float As[TILE][LDSS];
    __shared__ float Bs[TILE][LDSS];

    const float* Asrc = Cmod + ((size_t)bh * LSEQ + (size_t)ti * TILE) * DH;
    const float* Bsrc = Binv + ((size_t)bh * LSEQ + (size_t)tj * TILE) * DH;

    // 128 rows x 64 cols = 2048 float4 per matrix; 8 per thread
#pragma unroll
    for (int it = 0; it < 8; ++it) {
        int idx = it * 256 + tid;            // 0..2047
        int r   = idx >> 4;                  // /16
        int c   = (idx & 15) << 2;
        float4 va = *(const float4*)(Asrc + (size_t)r * DH + c);
        float4 vb = *(const float4*)(Bsrc + (size_t)r * DH + c);
        *(float4*)(&As[r][c]) = va;
        *(float4*)(&Bs[r][c]) = vb;
    }
    __syncthreads();

    const int wave = tid >> 5;
    const int lane = tid & 31;
    const int half = lane >> 4;
    const int lm   = lane & 15;
    const int rowBase = wave * 16;

    v8f zero8 = {0.f, 0.f, 0.f, 0.f, 0.f, 0.f, 0.f, 0.f};
    v8f acc[8];
#pragma unroll
    for (int n = 0; n < 8; ++n) acc[n] = zero8;

#pragma unroll
    for (int kk = 0; kk < 16; ++kk) {
        const int k0 = kk * 4 + half * 2;
        // A fragment (16x4 f32): lane m, VGPR0/1 = K = 2*half + {0,1}
        v2f a = *(const v2f*)(&As[rowBase + lm][k0]);
#pragma unroll
        for (int n = 0; n < 8; ++n) {
            // B fragment (4x16 f32): lane n holds column n, same K packing
            v2f bfr = *(const v2f*)(&Bs[n * 16 + lm][k0]);
            acc[n] = __builtin_amdgcn_wmma_f32_16x16x4_f32(
                false, a, false, bfr, (short)0, acc[n], false, false);
        }
    }

    const int rowG = ti * TILE + rowBase;
    const int colG = tj * TILE;

    if (ti != tj) {
        // strictly-lower tile: store everything
#pragma unroll
        for (int n = 0; n < 8; ++n) {
            int j = colG + n * 16 + lm;
#pragma unroll
            for (int v = 0; v < 8; ++v) {
                int i = rowG + v + 8 * half;
                outT[(size_t)i * LSEQ + j] = acc[n][v];
            }
        }
    } else {
        // diagonal tile: mask strict upper, override diagonal with C.B
        const float* dg = diag + (size_t)bh * LSEQ;
#pragma unroll
        for (int n = 0; n < 8; ++n) {
            int j = colG + n * 16 + lm;
#pragma unroll
            for (int v = 0; v < 8; ++v) {
                int i = rowG + v + 8 * half;
                float val = acc[n][v];
                if (j > i)      val = 0.0f;
                else if (j == i) val = dg[i];
                outT[(size_t)i * LSEQ + j] = val;
            }
        }
    }
}

// ---------------------------------------------------------------------------
// Launch
// ---------------------------------------------------------------------------
extern "C" void kernel_launch(void* const* d_in, const int* in_sizes, int n_in,
                              void* d_out, int out_size, void* d_ws, size_t ws_size,
                              hipStream_t stream) {
    const float* B = (const float*)d_in[0];
    const float* C = (const float*)d_in[1];
    const float* S = (const float*)d_in[2];
    float* out = (float*)d_out;

    // workspace layout: Cmod (2M floats) | Binv (2M floats) | diag (32K floats)
    float* Cmod = (float*)d_ws;
    float* Binv = Cmod + (size_t)NBH * LSEQ * DH;
    float* diag = Binv + (size_t)NBH * LSEQ * DH;

    tmwg_scan_kernel<<<(NBH * DH + 127) / 128, 128, 0, stream>>>(B, C, S, Cmod, Binv);
    tmwg_diag_kernel<<<(NBH * LSEQ + 255) / 256, 256, 0, stream>>>(B, C, diag);

    dim3 grid(LSEQ / TILE, LSEQ / TILE, NBH);   // (tj, ti, bh) = 16x16x16
    tmwg_gemm_kernel<<<grid, 256, 0, stream>>>(Cmod, Binv, diag, out);
}